// ChamferLoss_29068338659681
// MI455X (gfx1250) — compile-verified
//
#include <hip/hip_runtime.h>
#include <hip/hip_bf16.h>

typedef float v2f __attribute__((ext_vector_type(2)));
typedef float v8f __attribute__((ext_vector_type(8)));

#define TILE 16

// ---------------------------------------------------------------------------
// Build WMMA-fragment-ready records. pc is [B,3,N] channel-major.
// For each 16-point tile ct we emit 32 records (one per wave lane):
//   lane 0-15 : { x[ct*16+lane],   y[ct*16+lane],    |p|^2, 0 }   (k=0,1)
//   lane 16-31: { z[ct*16+lane-16], 0,               |p|^2, 0 }   (k=2,3)
// The hot loop then needs ONE global_load_b128 and no lane-half selects:
// bv = {rec.x, rec.y}, norm = rec.z. Same layout serves A and B fragments.
// ---------------------------------------------------------------------------
__global__ void chamfer_pack_kernel(const float* __restrict__ pc,
                                    float4* __restrict__ out,
                                    int nrec, int N) {
    int i = blockIdx.x * blockDim.x + threadIdx.x;
    if (i >= nrec) return;
    int recPerBatch = N * 2;              // (N/16) tiles * 32 records
    int b    = i / recPerBatch;
    int rem  = i - b * recPerBatch;
    int ct   = rem >> 5;
    int lane = rem & 31;
    int n    = ct * TILE + (lane & 15);
    const float* p = pc + (size_t)b * 3 * N;
    float x = p[n];
    float y = p[N + n];
    float z = p[2 * N + n];
    float nrm = x * x + y * y + z * z;
    out[i] = (lane < 16) ? make_float4(x, y, nrm, 0.0f)
                         : make_float4(z, 0.0f, nrm, 0.0f);
}

// ---------------------------------------------------------------------------
// Nearest-neighbor min-distance, one direction, from fragment-ready records.
//   distOut[b*N + m] = min_n ( |Q_m|^2 + |T_n|^2 - 2 Q_m . T_n )
// One wave = one 16-row query tile; loop over all 16-col target tiles,
// one V_WMMA_F32_16X16X4_F32 per tile (K padded 3->4 with zeros).
// C/D layout (16x16 f32, 8 VGPRs): lane holds column n = lane%16;
//   VGPR v holds row m = v (lanes 0-15) or v+8 (lanes 16-31).
// ---------------------------------------------------------------------------
__global__ void __launch_bounds__(256)
chamfer_nn_kernel(const float4* __restrict__ Fq,   // query frags  [B*N*2]
                  const float4* __restrict__ Ft,   // target frags [B*N*2]
                  float* __restrict__ distOut,     // [B*N]
                  int N) {
    const int lane = threadIdx.x & 31;
    const int warp = threadIdx.x >> 5;
    const int tilesPerBatch = N / TILE;

    int task = blockIdx.x * (blockDim.x >> 5) + warp;   // one row-tile / wave
    int b    = task / tilesPerBatch;
    int rt   = task - b * tilesPerBatch;

    const float4* Q = Fq + (size_t)b * N * 2;
    const float4* T = Ft + (size_t)b * N * 2;

    // --- A fragment: one b128 load, fragment-ready, reused all iterations ---
    float4 qa = Q[rt * 32 + lane];
    v2f av;
    av.x = qa.x;
    av.y = qa.y;

    const float INF = __builtin_huge_valf();
    float rm[8];
#pragma unroll
    for (int v = 0; v < 8; ++v) rm[v] = INF;

#pragma unroll 4
    for (int ct = 0; ct < tilesPerBatch; ++ct) {
        // --- B fragment + norm: ONE b128 load, no shuffling needed ---
        float4 tb = T[ct * 32 + lane];
        v2f bv;
        bv.x = tb.x;
        bv.y = tb.y;
        float nb = tb.z;               // |T_n|^2, same n in both lane halves

        v8f c = {};
        // D = A(16x4) * B(4x16) + 0  -> v_wmma_f32_16x16x4_f32
        c = __builtin_amdgcn_wmma_f32_16x16x4_f32(
                /*neg_a=*/false, av,
                /*neg_b=*/false, bv,
                /*c_mod=*/(short)0, c,
                /*reuse_a=*/false, /*reuse_b=*/false);

        // partial distance (|Q|^2 added after the min): nb - 2*dot
#pragma unroll
        for (int v = 0; v < 8; ++v) {
            float d = fmaf(-2.0f, c[v], nb);
            rm[v] = fminf(rm[v], d);
        }
    }

    // --- row-min across the 16 lanes of each half-wave, then add |Q_m|^2 ---
    const bool hi      = lane >= 16;
    const int  l16     = lane & 15;
    const int  rowbase = rt * TILE;
#pragma unroll
    for (int v = 0; v < 8; ++v) {
        float r = rm[v];
        r = fminf(r, __shfl_xor(r, 8, 16));
        r = fminf(r, __shfl_xor(r, 4, 16));
        r = fminf(r, __shfl_xor(r, 2, 16));
        r = fminf(r, __shfl_xor(r, 1, 16));
        if (l16 == 0) {
            int dm = v + (hi ? 8 : 0);             // row within tile, 0..15
            float qn = Q[rt * 32 + dm].z;          // |Q_m|^2 (low-half record)
            distOut[(size_t)b * N + rowbase + dm] = r + qn;
        }
    }
}

// ---------------------------------------------------------------------------
// Final mean: out = sum(d1 + d2) / (2 * total)   (total = B*N)
// ---------------------------------------------------------------------------
__global__ void chamfer_reduce_kernel(const float* __restrict__ d1,
                                      const float* __restrict__ d2,
                                      float* __restrict__ out,
                                      int total) {
    __shared__ float s[256];
    float acc = 0.0f;
    for (int i = threadIdx.x; i < total; i += blockDim.x)
        acc += d1[i] + d2[i];
    s[threadIdx.x] = acc;
    __syncthreads();
    for (int w = 128; w > 0; w >>= 1) {
        if ((int)threadIdx.x < w) s[threadIdx.x] += s[threadIdx.x + w];
        __syncthreads();
    }
    if (threadIdx.x == 0) out[0] = s[0] / (2.0f * (float)total);
}

// ---------------------------------------------------------------------------
extern "C" void kernel_launch(void* const* d_in, const int* in_sizes, int n_in,
                              void* d_out, int out_size, void* d_ws, size_t ws_size,
                              hipStream_t stream) {
    const float* pc1 = (const float*)d_in[0];   // [B,3,N]
    const float* pc2 = (const float*)d_in[1];   // [B,3,N]

    const int B = 2;
    const int N = in_sizes[0] / (B * 3);        // 16384
    const int total = B * N;
    const int nrec  = B * N * 2;                // fragment records per cloud

    float*  ws    = (float*)d_ws;
    float4* f1    = (float4*)ws;                    // nrec float4
    float4* f2    = (float4*)(ws + 4 * nrec);       // nrec float4
    float*  dist1 = ws + 8 * nrec;                  // total floats
    float*  dist2 = ws + 8 * nrec + total;          // total floats

    // 1) build fragment-ready records {v0, v1, |p|^2, 0}
    {
        dim3 blk(256), grd((nrec + 255) / 256);
        chamfer_pack_kernel<<<grd, blk, 0, stream>>>(pc1, f1, nrec, N);
        chamfer_pack_kernel<<<grd, blk, 0, stream>>>(pc2, f2, nrec, N);
    }

    // 2) NN min distances, both directions (8 waves/block, 1 row-tile/wave)
    {
        int tasks = B * (N / TILE);                 // 2048 wave-tasks
        dim3 blk(256), grd(tasks / 8);
        chamfer_nn_kernel<<<grd, blk, 0, stream>>>(f1, f2, dist1, N);
        chamfer_nn_kernel<<<grd, blk, 0, stream>>>(f2, f1, dist2, N);
    }

    // 3) mean
    chamfer_reduce_kernel<<<1, 256, 0, stream>>>(dist1, dist2, (float*)d_out, total);
}